// CrossTuckerLayer_33079838114298
// MI455X (gfx1250) — compile-verified
//
#include <hip/hip_runtime.h>

typedef __attribute__((ext_vector_type(2))) float v2f;
typedef __attribute__((ext_vector_type(4))) float v4f;
typedef __attribute__((ext_vector_type(8))) float v8f;

#define NSAMPLES 2048
#define XPER     16384   // 16*16*64 floats per sample
#define OUTPER   32768   // 256*128 floats per sample

__launch_bounds__(256, 2)
__global__ void CrossTuckerLayer_kernel(const float* __restrict__ x,
                                        const float* __restrict__ core,
                                        const float* __restrict__ u0,
                                        const float* __restrict__ u1,
                                        const float* __restrict__ u2,
                                        const float* __restrict__ a0,
                                        const float* __restrict__ a1,
                                        const float* __restrict__ a2,
                                        float* __restrict__ out)
{
    __shared__ float sG[256 * 4];   // G[row][AB], row = o*16+p, AB = A*2+B
    __shared__ float sW[128 * 4];   // W^T[q][AB]
    __shared__ float wpart[8 * 8];  // per-wave partial t
    __shared__ float sred[8];       // reduced t[D,E,F]

    const int tid    = threadIdx.x;
    const int wave   = tid >> 5;
    const int lane   = tid & 31;
    const int sample = blockIdx.x;

    const float* xs   = x + (size_t)sample * XPER;
    float*       outs = out + (size_t)sample * OUTPER;

    // ---- Step 1: e-contraction, one (a,d) row of x per thread ----
    const int ai = tid >> 4;   // a index (also o for G build)
    const int di = tid & 15;   // d index (also p for G build)
    const float* xr = xs + tid * 64;

    float pe0 = 0.f, pe1 = 0.f;
    #pragma unroll
    for (int e4 = 0; e4 < 16; ++e4) {
        v4f xv = *(const v4f*)(xr + e4 * 4);
        #pragma unroll
        for (int j = 0; j < 4; ++j) {
            const int e  = e4 * 4 + j;
            const float xe = xv[j];
            pe0 = fmaf(xe, a2[e * 2 + 0], pe0);
            pe1 = fmaf(xe, a2[e * 2 + 1], pe1);
        }
    }

    float pt[8];
    {
        const float a00 = a0[ai * 2 + 0], a01 = a0[ai * 2 + 1];
        const float a10 = a1[di * 2 + 0], a11 = a1[di * 2 + 1];
        pt[0] = a00 * a10 * pe0;  pt[1] = a00 * a10 * pe1;   // D=0,E=0,F=0/1
        pt[2] = a00 * a11 * pe0;  pt[3] = a00 * a11 * pe1;   // D=0,E=1
        pt[4] = a01 * a10 * pe0;  pt[5] = a01 * a10 * pe1;   // D=1,E=0
        pt[6] = a01 * a11 * pe0;  pt[7] = a01 * a11 * pe1;   // D=1,E=1
    }

    // Deterministic wave32 tree reduction, then cross-wave sum in fixed order
    #pragma unroll
    for (int k = 0; k < 8; ++k) {
        float v = pt[k];
        #pragma unroll
        for (int off = 16; off > 0; off >>= 1) v += __shfl_down(v, off, 32);
        pt[k] = v;
    }
    if (lane == 0) {
        #pragma unroll
        for (int k = 0; k < 8; ++k) wpart[wave * 8 + k] = pt[k];
    }
    __syncthreads();
    if (tid < 8) {
        float s = 0.f;
        #pragma unroll
        for (int w = 0; w < 8; ++w) s += wpart[w * 8 + tid];
        sred[tid] = s;
    }
    __syncthreads();

    float t[8];
    #pragma unroll
    for (int k = 0; k < 8; ++k) t[k] = sred[k];

    // ---- Step 2: s[A,B,C] = core . t (redundant per-thread, cached reads) ----
    float sv[8];
    #pragma unroll
    for (int abc = 0; abc < 8; ++abc) {
        float acc = 0.f;
        #pragma unroll
        for (int def = 0; def < 8; ++def)
            acc = fmaf(core[abc * 8 + def], t[def], acc);
        sv[abc] = acc;
    }

    // ---- Build LDS operands for rank-4 expansion GEMM ----
    // G[row=tid][AB] = u0[o,A]*u1[p,B]  (sample-invariant, cheap to rebuild)
    {
        const float u00 = u0[ai * 2 + 0], u01 = u0[ai * 2 + 1];
        const float u10 = u1[di * 2 + 0], u11 = u1[di * 2 + 1];
        sG[tid * 4 + 0] = u00 * u10;
        sG[tid * 4 + 1] = u00 * u11;
        sG[tid * 4 + 2] = u01 * u10;
        sG[tid * 4 + 3] = u01 * u11;
    }
    // W^T[q][AB] = sv[AB*2+0]*u2[q,0] + sv[AB*2+1]*u2[q,1]
    {
        const int q    = tid & 127;
        const int half = tid >> 7;
        const float uq0 = u2[q * 2 + 0], uq1 = u2[q * 2 + 1];
        #pragma unroll
        for (int j = 0; j < 2; ++j) {
            const int ab = half * 2 + j;
            sW[q * 4 + ab] = sv[ab * 2 + 0] * uq0 + sv[ab * 2 + 1] * uq1;
        }
    }
    __syncthreads();

    // ---- Step 3: [256x4] x [4x128] -> [256x128] via v_wmma_f32_16x16x4_f32 ----
    // 128 output tiles of 16x16; 16 tiles per wave; K=4 -> single WMMA per tile.
    const int sub = lane & 15;   // M (for A/D) or N (for B/D)
    const int hi  = lane >> 4;   // 0: K=0,1 ; 1: K=2,3  (and D rows M+0 vs M+8)
    #pragma unroll
    for (int ti = 0; ti < 16; ++ti) {
        const int tileId = wave * 16 + ti;
        const int r = tileId >> 3;   // row tile 0..15
        const int c = tileId & 7;    // q   tile 0..7

        // A fragment: A[M = r*16+sub, K = hi*2 + {0,1}]  (one ds_read_b64)
        v2f afrag = *(const v2f*)&sG[(r * 16 + sub) * 4 + hi * 2];
        // B fragment: B[K = hi*2 + {0,1}, N = c*16+sub]  (one ds_read_b64)
        v2f bfrag = *(const v2f*)&sW[(c * 16 + sub) * 4 + hi * 2];

        v8f cfrag = {};
        v8f dfrag = __builtin_amdgcn_wmma_f32_16x16x4_f32(
            /*neg_a=*/false, afrag, /*neg_b=*/false, bfrag,
            /*c_mod=*/(short)0, cfrag, /*reuse_a=*/false, /*reuse_b=*/false);

        // D: vgpr j holds row M = r*16 + hi*8 + j, col N = c*16 + sub
        float* ob = outs + (size_t)(c * 16 + sub);
        const int rowbase = r * 16 + hi * 8;
        #pragma unroll
        for (int j = 0; j < 8; ++j) {
            ob[(size_t)(rowbase + j) * 128] = dfrag[j];
        }
    }
}

extern "C" void kernel_launch(void* const* d_in, const int* in_sizes, int n_in,
                              void* d_out, int out_size, void* d_ws, size_t ws_size,
                              hipStream_t stream) {
    (void)in_sizes; (void)n_in; (void)out_size; (void)d_ws; (void)ws_size;
    const float* x    = (const float*)d_in[0];
    const float* core = (const float*)d_in[1];
    const float* u0   = (const float*)d_in[2];
    const float* u1   = (const float*)d_in[3];
    const float* u2   = (const float*)d_in[4];
    const float* a0   = (const float*)d_in[5];
    const float* a1   = (const float*)d_in[6];
    const float* a2   = (const float*)d_in[7];
    float* out = (float*)d_out;

    dim3 grid(NSAMPLES);
    dim3 block(256);
    hipLaunchKernelGGL(CrossTuckerLayer_kernel, grid, block, 0, stream,
                       x, core, u0, u1, u2, a0, a1, a2, out);
}